// Custom_Model_Quant_Embedding_Group_7842610282551
// MI455X (gfx1250) — compile-verified
//
#include <hip/hip_runtime.h>

// CDNA5 / gfx1250 quantized embedding-bag + concat.
// Per 4 gathered rows: V_WMMA_F32_16X16X4_F32 with
//   A[16x4]  = the 4 per-row scales broadcast down M (exact f32)
//   B[4x16]  = one 16-dim tile of the 4 int4-code rows (cvt to f32)
//   D[16x16] row 0 = running bag sum for that dim tile; 4 accumulators = 64 dims.
// HBM-gather bound (~216 MB of random row reads -> ~9us floor at 23.3 TB/s);
// the main loop is branch-free and 2x group-unrolled for load pipelining.

typedef float v2f __attribute__((ext_vector_type(2)));
typedef float v8f __attribute__((ext_vector_type(8)));

#define EMB_DIM 64

// One group of 4 indices -> 4 WMMAs (one per 16-dim tile).
// v1/v2/v3 are validity masks; callers pass compile-time `true` in the hot
// loop so every cndmask folds away. Loads are always in-bounds (indices are
// pre-clamped by the caller); masking zeroes the scale/bias contribution only.
__device__ __forceinline__ void group4(const int* __restrict__ wq,
                                       const float* __restrict__ scales,
                                       const float* __restrict__ biases,
                                       long long i0, long long i1,
                                       long long i2, long long i3,
                                       bool v1, bool v2, bool v3,
                                       bool lo, int col,
                                       v8f& acc0, v8f& acc1, v8f& acc2, v8f& acc3,
                                       float& bsum)
{
    // A (32-bit 16x4): VGPR0 = K0 (lanes 0-15) / K2 (lanes 16-31),
    //                  VGPR1 = K1 (lanes 0-15) / K3 (lanes 16-31).
    const long long ia_x = lo ? i0 : i2;
    const long long ia_y = lo ? i1 : i3;
    const bool      va_x = lo ? true : v2;
    const bool      va_y = lo ? v1   : v3;
    const float sx = scales[ia_x], sy = scales[ia_y];   // unconditional loads
    const float bx = biases[ia_x], by = biases[ia_y];
    v2f a;
    a.x = va_x ? sx : 0.0f;                             // cndmask, no branches
    a.y = va_y ? sy : 0.0f;
    bsum += va_x ? bx : 0.0f;                           // lanes 0-15: b0+b1
    bsum += va_y ? by : 0.0f;                           // lanes 16-31: b2+b3

    // B (32-bit 4x16): VGPR0 = row K0 (lanes 0-15) / K1 (lanes 16-31),
    //                  VGPR1 = row K2 (lanes 0-15) / K3 (lanes 16-31).
    const long long rb_x = lo ? i0 : i1;
    const long long rb_y = lo ? i2 : i3;
    const int* px = wq + rb_x * EMB_DIM + col;
    const int* py = wq + rb_y * EMB_DIM + col;
    v2f b0v, b1v, b2v, b3v;
    b0v.x = (float)px[0];   b0v.y = (float)py[0];
    b1v.x = (float)px[16];  b1v.y = (float)py[16];
    b2v.x = (float)px[32];  b2v.y = (float)py[32];
    b3v.x = (float)px[48];  b3v.y = (float)py[48];

    acc0 = __builtin_amdgcn_wmma_f32_16x16x4_f32(false, a, false, b0v, (short)0, acc0, false, false);
    acc1 = __builtin_amdgcn_wmma_f32_16x16x4_f32(false, a, false, b1v, (short)0, acc1, false, false);
    acc2 = __builtin_amdgcn_wmma_f32_16x16x4_f32(false, a, false, b2v, (short)0, acc2, false, false);
    acc3 = __builtin_amdgcn_wmma_f32_16x16x4_f32(false, a, false, b3v, (short)0, acc3, false, false);
}

__global__ __launch_bounds__(256)
void qemb_bag_wmma_kernel(const int* __restrict__ wq,
                          const float* __restrict__ scales,
                          const float* __restrict__ biases,
                          const long long* __restrict__ indices,
                          const long long* __restrict__ offsets,
                          const float* __restrict__ cat,
                          float* __restrict__ out,
                          int num_bags, int cat_dim)
{
    const int wave = threadIdx.x >> 5;
    const int lane = threadIdx.x & 31;
    const int bag  = blockIdx.x * (blockDim.x >> 5) + wave;
    if (bag >= num_bags) return;              // wave-uniform

    const long long start = offsets[bag];
    const long long end   = offsets[bag + 1];
    const int n = (int)(end - start);

    const bool lo  = lane < 16;
    const int  col = lane & 15;
    const long long* __restrict__ ibase = indices + start;

    v8f acc0 = {}, acc1 = {}, acc2 = {}, acc3 = {};
    float bsum = 0.0f;

    int it = 0;
    // ---- hot loop: 8 indices / iteration, fully unconditional loads ----
    for (; it + 8 <= n; it += 8) {
        __builtin_prefetch(ibase + it + 8, 0, 1);      // global_prefetch_b8
        const long long i0 = ibase[it + 0], i1 = ibase[it + 1];
        const long long i2 = ibase[it + 2], i3 = ibase[it + 3];
        const long long i4 = ibase[it + 4], i5 = ibase[it + 5];
        const long long i6 = ibase[it + 6], i7 = ibase[it + 7];
        group4(wq, scales, biases, i0, i1, i2, i3, true, true, true,
               lo, col, acc0, acc1, acc2, acc3, bsum);
        group4(wq, scales, biases, i4, i5, i6, i7, true, true, true,
               lo, col, acc0, acc1, acc2, acc3, bsum);
    }
    // ---- tail: branch-free masked groups of 4 (clamped offsets) ----
    for (; it < n; it += 4) {
        const int nm1 = n - 1;
        const long long i0 = ibase[it];
        const long long i1 = ibase[(it + 1 < nm1) ? it + 1 : nm1];
        const long long i2 = ibase[(it + 2 < nm1) ? it + 2 : nm1];
        const long long i3 = ibase[(it + 3 < nm1) ? it + 3 : nm1];
        group4(wq, scales, biases, i0, i1, i2, i3,
               (it + 1) < n, (it + 2) < n, (it + 3) < n,
               lo, col, acc0, acc1, acc2, acc3, bsum);
    }

    // combine half-wave bias partial sums (wave32 cross-half swap)
    bsum += __shfl_xor(bsum, 16, 32);

    float* orow = out + (long long)bag * (2 * EMB_DIM + cat_dim);

    // D row 0 (M=0) lives in VGPR0 of lanes 0-15 -> acc[0] for lane<16.
    if (lo) {
        const float e0 = acc0[0] + bsum;
        const float e1 = acc1[0] + bsum;
        const float e2 = acc2[0] + bsum;
        const float e3 = acc3[0] + bsum;
        // eb1 == eb2 in the reference -> write the 64 dims twice
        orow[ 0 + col] = e0;  orow[EMB_DIM +  0 + col] = e0;
        orow[16 + col] = e1;  orow[EMB_DIM + 16 + col] = e1;
        orow[32 + col] = e2;  orow[EMB_DIM + 32 + col] = e2;
        orow[48 + col] = e3;  orow[EMB_DIM + 48 + col] = e3;
    }

    // concat cat_input (16B-aligned) with float4 copies
    const float4* crow = (const float4*)(cat + (long long)bag * cat_dim);
    float4*       drow = (float4*)(orow + 2 * EMB_DIM);
    for (int j = lane; j < (cat_dim >> 2); j += 32) {
        drow[j] = crow[j];
    }
}

extern "C" void kernel_launch(void* const* d_in, const int* in_sizes, int n_in,
                              void* d_out, int out_size, void* d_ws, size_t ws_size,
                              hipStream_t stream) {
    const int*       wq      = (const int*)d_in[0];        // weights_q  int32
    const float*     scales  = (const float*)d_in[1];      // float32
    const float*     biases  = (const float*)d_in[2];      // float32
    const long long* indices = (const long long*)d_in[3];  // int64
    const long long* offsets = (const long long*)d_in[4];  // int64
    const float*     cat     = (const float*)d_in[5];      // float32
    float*           outp    = (float*)d_out;              // float32 output

    const int num_bags = in_sizes[4] - 1;                  // offsets = BAGS+1
    const int cat_dim  = in_sizes[5] / num_bags;

    const int waves_per_block = 8;                         // 256 threads, wave32
    dim3 block(32 * waves_per_block);
    dim3 grid((num_bags + waves_per_block - 1) / waves_per_block);

    hipLaunchKernelGGL(qemb_bag_wmma_kernel, grid, block, 0, stream,
                       wq, scales, biases, indices, offsets, cat, outp,
                       num_bags, cat_dim);
}